// PatternSupervisedLoss_46523085750361
// MI455X (gfx1250) — compile-verified
//
#include <hip/hip_runtime.h>
#include <math.h>

#define BATCH   16384
#define DIM     1024
#define NBINS   100
#define NPAT    10
#define NCLS    1000
#define PADCAP  (BATCH + NBINS * 16)   // 17984: bin-padded order list capacity
#define MAXG    (PADCAP / 16)          // 1124 groups of 16 samples
#define TEMP_INV 10.0f                 // 1 / TEMP

typedef float v2f __attribute__((ext_vector_type(2)));
typedef float v8f __attribute__((ext_vector_type(8)));

// ---------------- bucketing ----------------

__global__ void init_kernel(int* counts, int* group_bin, int* order_list) {
    int i = blockIdx.x * blockDim.x + threadIdx.x;
    if (i < NBINS)  counts[i] = 0;
    if (i < MAXG)   group_bin[i] = -1;
    if (i < PADCAP) order_list[i] = -1;
}

__device__ __forceinline__ int bin_of(int t) {
    int tc = t / 10;
    if (tc < 0) tc = 0;
    if (tc > NBINS - 1) tc = NBINS - 1;
    return tc;
}

__global__ void count_kernel(const int* __restrict__ ts, int* counts) {
    int b = blockIdx.x * blockDim.x + threadIdx.x;
    if (b < BATCH) atomicAdd(&counts[bin_of(ts[b])], 1);
}

// single-thread: bin regions padded to multiples of 16 -> group g owns slots [16g,16g+16)
__global__ void scan_kernel(const int* __restrict__ counts, int* cursor, int* group_bin) {
    if (threadIdx.x == 0 && blockIdx.x == 0) {
        int g = 0;
        for (int j = 0; j < NBINS; ++j) {
            cursor[j] = g * 16;
            int ng = (counts[j] + 15) / 16;
            for (int t = 0; t < ng; ++t) group_bin[g++] = j;
        }
    }
}

__global__ void scatter_kernel(const int* __restrict__ ts, int* cursor, int* order_list) {
    int b = blockIdx.x * blockDim.x + threadIdx.x;
    if (b < BATCH) {
        int pos = atomicAdd(&cursor[bin_of(ts[b])], 1);
        order_list[pos] = b;
    }
}

// ---------------- centroid normalize -> K-major padded B layout ----------------
// cnT[bin][k][p16] with p=10..15 zero; matches WMMA B fragment addressing.

__global__ void centroid_kernel(const float* __restrict__ CC, float* __restrict__ cnT) {
    int bin = blockIdx.x >> 4;
    int p   = blockIdx.x & 15;
    int tid = threadIdx.x;                 // 128 threads
    float* dst = cnT + (size_t)bin * DIM * 16;
    if (p >= NPAT) {
        for (int k = tid; k < DIM; k += 128) dst[(size_t)k * 16 + p] = 0.0f;
        return;
    }
    const float* row = CC + ((size_t)bin * NPAT + p) * DIM;
    __shared__ float red[128];
    float q = 0.0f;
    for (int k = tid; k < DIM; k += 128) { float v = row[k]; q += v * v; }
    red[tid] = q;
    __syncthreads();
    for (int s = 64; s > 0; s >>= 1) {
        if (tid < s) red[tid] += red[tid + s];
        __syncthreads();
    }
    float inv = 1.0f / fmaxf(sqrtf(red[0]), 1e-8f);
    for (int k = tid; k < DIM; k += 128) dst[(size_t)k * 16 + p] = row[k] * inv;
}

// ---------------- WMMA sims kernel ----------------
// Group g: 16 samples sharing one bin. 4 waves split K=1024 into quarters of 256.
// A (16x4 f32): lanes0-15 {v0=A[m][k0],v1=A[m][k0+1]}, lanes16-31 {k0+2,k0+3}
// B (4x16 f32): lanes0-15 {v0=B[k0][n],v1=B[k0+1][n]}, lanes16-31 {k0+2,k0+3}
// D (16x16 f32): vgpr i: lanes0-15 M=i, lanes16-31 M=i+8; N=lane&15
//
// Sentinel slots (samp == -1) load sample row 0 unconditionally: their D rows,
// ssq and outputs are never consumed (post-processing re-checks order_list),
// and this keeps the inner loop free of EXEC-mask branches around the A load.

__global__ void __launch_bounds__(128)
wmma_sims_kernel(const float* __restrict__ SF, const float* __restrict__ cnT,
                 const int* __restrict__ order_list, const int* __restrict__ group_bin,
                 float* __restrict__ best, float* __restrict__ pp) {
    int g = blockIdx.x;
    int bin = group_bin[g];
    if (bin < 0) return;

    int tid  = threadIdx.x;
    int wave = tid >> 5;
    int lane = tid & 31;
    int half = lane >> 4;
    int r    = lane & 15;

    int samp = order_list[g * 16 + r];
    int srow = (samp >= 0) ? samp : 0;          // safe row for sentinel lanes
    const float* arow = SF + (size_t)srow * DIM;
    const float* bb   = cnT + (size_t)bin * DIM * 16;

    v8f acc = {0.f, 0.f, 0.f, 0.f, 0.f, 0.f, 0.f, 0.f};
    float ssq = 0.0f;
    int kb = wave * 256;
    for (int k0 = kb; k0 < kb + 256; k0 += 4) {
        int ka = k0 + half * 2;
        v2f a = *(const v2f*)(arow + ka);       // branch-free A fragment
        ssq += a.x * a.x + a.y * a.y;
        v2f bv;
        bv.x = bb[(size_t)ka * 16 + r];
        bv.y = bb[(size_t)(ka + 1) * 16 + r];
        acc = __builtin_amdgcn_wmma_f32_16x16x4_f32(
            /*neg_a=*/false, a, /*neg_b=*/false, bv,
            /*c_mod=*/(short)0, acc, /*reuse_a=*/false, /*reuse_b=*/false);
    }
    ssq += __shfl_xor(ssq, 16);   // lane L + L+16 cover all k mod 4

    __shared__ float sPart[4][16][16];
    __shared__ float qPart[4][16];
    if (lane < 16) qPart[wave][r] = ssq;
    #pragma unroll
    for (int i = 0; i < 8; ++i) {
        int m = i + half * 8;
        sPart[wave][m][r] = acc[i];
    }
    __syncthreads();

    if (tid < 16) {
        int m  = tid;
        int sm = order_list[g * 16 + m];
        if (sm >= 0) {
            float q = qPart[0][m] + qPart[1][m] + qPart[2][m] + qPart[3][m];
            float inv = 1.0f / fmaxf(sqrtf(q), 1e-8f);
            float s[NPAT];
            float bst = -INFINITY;
            #pragma unroll
            for (int n = 0; n < NPAT; ++n) {
                float v = (sPart[0][m][n] + sPart[1][m][n] +
                           sPart[2][m][n] + sPart[3][m][n]) * inv;
                s[n] = v;
                bst = fmaxf(bst, v);
            }
            float e[NPAT];
            float se = 0.0f;
            #pragma unroll
            for (int n = 0; n < NPAT; ++n) { e[n] = expf((s[n] - bst) * TEMP_INV); se += e[n]; }
            float invse = 1.0f / se;
            best[sm] = bst;
            #pragma unroll
            for (int n = 0; n < NPAT; ++n) pp[(size_t)sm * NPAT + n] = e[n] * invse;
        }
    }
}

// ---------------- per-sample logits: logsumexp + slice CE + entropy ----------------

__global__ void __launch_bounds__(256)
logits_kernel(const float* __restrict__ SL, const int* __restrict__ ts,
              const float* __restrict__ pp, float* __restrict__ ce, float* __restrict__ ent) {
    int wave = threadIdx.x >> 5;
    int lane = threadIdx.x & 31;
    int b = blockIdx.x * 8 + wave;
    if (b >= BATCH) return;
    const float* row = SL + (size_t)b * NCLS;

    float m = -INFINITY;
    for (int j = lane; j < NCLS; j += 32) m = fmaxf(m, row[j]);
    for (int off = 16; off >= 1; off >>= 1) m = fmaxf(m, __shfl_xor(m, off));
    float ssum = 0.0f;
    for (int j = lane; j < NCLS; j += 32) ssum += expf(row[j] - m);
    for (int off = 16; off >= 1; off >>= 1) ssum += __shfl_xor(ssum, off);
    float logZ = m + logf(ssum);

    if (lane == 0) {
        int c0 = bin_of(ts[b]) * NPAT;
        float sl[NPAT];
        float smax = -INFINITY;
        #pragma unroll
        for (int i = 0; i < NPAT; ++i) { sl[i] = row[c0 + i]; smax = fmaxf(smax, sl[i]); }
        float Z2 = 0.0f;
        #pragma unroll
        for (int i = 0; i < NPAT; ++i) Z2 += expf(sl[i] - smax);
        float invZ2 = 1.0f / Z2;
        float ceb = 0.0f, entb = 0.0f;
        #pragma unroll
        for (int i = 0; i < NPAT; ++i) {
            float p = expf(sl[i] - smax) * invZ2;
            entb -= p * logf(p + 1e-8f);
            ceb  -= pp[(size_t)b * NPAT + i] * (sl[i] - logZ);
        }
        ce[b] = ceb;
        ent[b] = entb;
    }
}

// ---------------- final fixed-order reduction ----------------

__global__ void __launch_bounds__(256)
final_kernel(const float* __restrict__ best, const float* __restrict__ ce,
             const float* __restrict__ ent, float* __restrict__ out) {
    __shared__ float rb[256], rc[256], re[256];
    int tid = threadIdx.x;
    float sb = 0.0f, sc = 0.0f, se = 0.0f;
    for (int b = tid; b < BATCH; b += 256) { sb += best[b]; sc += ce[b]; se += ent[b]; }
    rb[tid] = sb; rc[tid] = sc; re[tid] = se;
    __syncthreads();
    for (int s = 128; s > 0; s >>= 1) {
        if (tid < s) { rb[tid] += rb[tid + s]; rc[tid] += rc[tid + s]; re[tid] += re[tid + s]; }
        __syncthreads();
    }
    if (tid == 0) {
        const float invB = 1.0f / (float)BATCH;
        float feat = 1.0f - rb[0] * invB;   // feature_sim_loss == rose_loss
        float cem  = rc[0] * invB;
        float entm = re[0] * invB;
        out[0] = 0.5f * feat + 0.3f * feat + 0.2f * cem + 0.05f * (-entm);
    }
}

// ---------------- launch ----------------

extern "C" void kernel_launch(void* const* d_in, const int* in_sizes, int n_in,
                              void* d_out, int out_size, void* d_ws, size_t ws_size,
                              hipStream_t stream) {
    (void)in_sizes; (void)n_in; (void)out_size; (void)ws_size;
    const float* SF = (const float*)d_in[0];      // student_features [B,D]
    // d_in[1] teacher_features: unused by reference
    const float* SL = (const float*)d_in[2];      // student_logits [B,1000]
    const float* CC = (const float*)d_in[3];      // crystal_centroids [100,10,D]
    const int*   TS = (const int*)d_in[4];        // timesteps [B]

    char* ws = (char*)d_ws;
    size_t o = 0;
    float* cnT       = (float*)(ws + o); o += (size_t)NBINS * DIM * 16 * sizeof(float);
    int*   counts    = (int*)  (ws + o); o += NBINS * sizeof(int);
    int*   cursor    = (int*)  (ws + o); o += NBINS * sizeof(int);
    int*   group_bin = (int*)  (ws + o); o += MAXG * sizeof(int);
    int*   order     = (int*)  (ws + o); o += PADCAP * sizeof(int);
    float* best      = (float*)(ws + o); o += (size_t)BATCH * sizeof(float);
    float* ce        = (float*)(ws + o); o += (size_t)BATCH * sizeof(float);
    float* ent       = (float*)(ws + o); o += (size_t)BATCH * sizeof(float);
    float* pp        = (float*)(ws + o); o += (size_t)BATCH * NPAT * sizeof(float);

    init_kernel   <<<(PADCAP + 255) / 256, 256, 0, stream>>>(counts, group_bin, order);
    count_kernel  <<<BATCH / 256, 256, 0, stream>>>(TS, counts);
    scan_kernel   <<<1, 1, 0, stream>>>(counts, cursor, group_bin);
    scatter_kernel<<<BATCH / 256, 256, 0, stream>>>(TS, cursor, order);
    centroid_kernel<<<NBINS * 16, 128, 0, stream>>>(CC, cnT);
    wmma_sims_kernel<<<MAXG, 128, 0, stream>>>(SF, cnT, order, group_bin, best, pp);
    logits_kernel <<<BATCH / 8, 256, 0, stream>>>(SL, TS, pp, ce, ent);
    final_kernel  <<<1, 256, 0, stream>>>(best, ce, ent, (float*)d_out);
}